// Pixelwise_7284264534030
// MI455X (gfx1250) — compile-verified
//
#include <hip/hip_runtime.h>
#include <hip/hip_bf16.h>
#include <math.h>

// ---------------- constants (match reference) ----------------
#define NS 10000
#define KF 3
static constexpr double SPEED_D  = 299792458.0 * 1000.0;    // mm/s
static constexpr double DMAX_D   = 10000.0;
static constexpr double FMAX_D   = SPEED_D / (2.0 * DMAX_D);
static constexpr double TAU_D    = 1.0 / FMAX_D;
static constexpr double PSRC_D   = 1.0e9;
static constexpr double PAMB_D   = 1.0e6;
static constexpr float  TAU_PSRC = (float)(TAU_D * PSRC_D);          // Mod rescale numerator
static constexpr float  AMB_COEF = (float)(PAMB_D * TAU_D);          // * kappa_k -> ambient term
static constexpr float  BSCALE   = (float)(1.0 / ((PAMB_D + PSRC_D) * TAU_D)); // GAMMA*BETA*T/TAU

// workspace layout (floats):
//  [0..3]       sumMod[k] (k<3), pad
//  [4..7]       amb[k] = P_AMB*kappa_k*TAU (k<3), amb[3]=0
//  [8..11]      mean[k]
//  [12..15]     invstd[k]
//  [16..40015]  CorrFs4  [NS][4]  (4th column = 0)
//  [40016..]    B0 [625*32]  (WMMA B-fragment VGPR0, per-lane order)
//  [60016..]    B1 [625*32]  (WMMA B-fragment VGPR1, per-lane order)
#define WS_SUMMOD 0
#define WS_AMB    4
#define WS_MEAN   8
#define WS_ISTD   12
#define WS_CORR   16
#define WS_B0     40016
#define WS_B1     60016

typedef __attribute__((ext_vector_type(2))) float v2f;
typedef __attribute__((ext_vector_type(8))) float v8f;

// ---------- A1: column sums of ModFs, kappas of clamped DemodFs ----------
__global__ __launch_bounds__(256) void colsum_kernel(const float* __restrict__ Mod,
                                                     const float* __restrict__ Dem,
                                                     float* __restrict__ ws) {
    __shared__ float r0[256], r1[256];
    const int tid = threadIdx.x;
    const int k = blockIdx.x;   // 0..2
    float sm = 0.f, sd = 0.f;
    for (int i = tid; i < NS; i += 256) {
        sm += Mod[i * KF + k];
        float d = Dem[i * KF + k];
        sd += fminf(fmaxf(d, 0.f), 1.f);
    }
    r0[tid] = sm; r1[tid] = sd;
    __syncthreads();
    for (int s = 128; s > 0; s >>= 1) {
        if (tid < s) { r0[tid] += r0[tid + s]; r1[tid] += r1[tid + s]; }
        __syncthreads();
    }
    if (tid == 0) {
        ws[WS_SUMMOD + k] = r0[0];
        ws[WS_AMB + k]    = AMB_COEF * (r1[0] / (float)NS);
        if (k == 0) { ws[WS_SUMMOD + 3] = 0.f; ws[WS_AMB + 3] = 0.f; }
    }
}

// ---------- A2: brute-force circular correlation -> CorrFs4 [NS][4] ----------
__global__ __launch_bounds__(256) void corr_kernel(const float* __restrict__ Mod,
                                                   const float* __restrict__ Dem,
                                                   float* __restrict__ ws) {
    __shared__ float demL[NS];   // 40 KB
    const int tid = threadIdx.x;
    const int n = blockIdx.x * 256 + tid;
    float* corr4 = ws + WS_CORR;
    for (int k = 0; k < KF; ++k) {
        __syncthreads();
        for (int i = tid; i < NS; i += 256) {
            float d = Dem[i * KF + k];
            demL[i] = fminf(fmaxf(d, 0.f), 1.f);
        }
        __syncthreads();
        if (n < NS) {
            const float sc = TAU_PSRC / ws[WS_SUMMOD + k];
            float acc = 0.f;
            const int split = NS - n;
            int t = 0;
            for (; t < split; ++t) acc = fmaf(Mod[t * KF + k], demL[n + t], acc);       // j = n+t
            for (; t < NS; ++t)    acc = fmaf(Mod[t * KF + k], demL[n + t - NS], acc);  // wrapped
            corr4[n * 4 + k] = sc * acc;
            if (k == 0) corr4[n * 4 + 3] = 0.f;
        }
    }
}

// ---------- A3: per-column mean / invstd (ddof=1) of CorrFs ----------
__global__ __launch_bounds__(256) void stats_kernel(float* __restrict__ ws) {
    __shared__ float red[256];
    __shared__ float meanSh;
    const float* corr4 = ws + WS_CORR;
    const int tid = threadIdx.x;
    for (int k = 0; k < KF; ++k) {
        float s = 0.f;
        for (int i = tid; i < NS; i += 256) s += corr4[i * 4 + k];
        __syncthreads();
        red[tid] = s; __syncthreads();
        for (int st = 128; st > 0; st >>= 1) { if (tid < st) red[tid] += red[tid + st]; __syncthreads(); }
        if (tid == 0) meanSh = red[0] / (float)NS;
        __syncthreads();
        const float mean = meanSh;
        float d = 0.f;
        for (int i = tid; i < NS; i += 256) { float x = corr4[i * 4 + k] - mean; d = fmaf(x, x, d); }
        __syncthreads();
        red[tid] = d; __syncthreads();
        for (int st = 128; st > 0; st >>= 1) { if (tid < st) red[tid] += red[tid + st]; __syncthreads(); }
        if (tid == 0) {
            ws[WS_MEAN + k] = mean;
            ws[WS_ISTD + k] = 1.0f / sqrtf(red[0] / (float)(NS - 1));
        }
        __syncthreads();
    }
}

// ---------- A4: build WMMA B-fragments of NormCorrFs in per-lane order ----------
// chunk c covers bins n = 16c..16c+15.
// B-VGPR0: lanes 0-15 -> K-row 0, lanes 16-31 -> K-row 2
// B-VGPR1: lanes 0-15 -> K-row 1, lanes 16-31 -> K-row 3 (zero pad)
__global__ __launch_bounds__(256) void build_b_kernel(float* __restrict__ ws) {
    const int g = blockIdx.x * 256 + threadIdx.x;
    if (g >= 625 * 32) return;
    const float* corr4 = ws + WS_CORR;
    const int c = g >> 5, l = g & 31;
    const int n = c * 16 + (l & 15);
    const int k0 = (l < 16) ? 0 : 2;
    ws[WS_B0 + g] = (corr4[n * 4 + k0] - ws[WS_MEAN + k0]) * ws[WS_ISTD + k0];
    ws[WS_B1 + g] = (l < 16) ? (corr4[n * 4 + 1] - ws[WS_MEAN + 1]) * ws[WS_ISTD + 1] : 0.f;
}

// ---------- main decode: WMMA scores + online exp-sum softmax-argmax ----------
__global__ __launch_bounds__(256) void decode_kernel(const float* __restrict__ gt,
                                                     const float* __restrict__ ws,
                                                     float* __restrict__ out) {
    const int lane = threadIdx.x & 31;
    const int wv   = threadIdx.x >> 5;
    const int base = blockIdx.x * 128 + wv * 16;   // 16 pixels per wave
    const float* corr4 = ws + WS_CORR;
    const float* B0 = ws + WS_B0;
    const float* B1 = ws + WS_B1;

    // --- per-pixel brightness + zero-norm (K=3, ddof=1) ---
    const int p = base + (lane & 15);
    const float depth = gt[p];                       // N_SAMP/D_MAX == 1.0 exactly
    int idx = ((int)rintf(depth)) % NS;              // round-half-even like jnp.round
    idx = (idx < 0) ? idx + NS : idx;
    const float4 cv = *(const float4*)(corr4 + idx * 4);
    const float b0v = BSCALE * (cv.x + ws[WS_AMB + 0]);
    const float b1v = BSCALE * (cv.y + ws[WS_AMB + 1]);
    const float b2v = BSCALE * (cv.z + ws[WS_AMB + 2]);
    const float m   = (b0v + b1v + b2v) * (1.f / 3.f);
    const float e0 = b0v - m, e1 = b1v - m, e2 = b2v - m;
    const float inv = 1.0f / sqrtf((e0 * e0 + e1 * e1 + e2 * e2) * 0.5f);  // ddof=1
    const float n0 = e0 * inv, n1 = e1 * inv, n2 = e2 * inv;

    // A fragment 16x4 f32: lane<16 -> {K0,K1}, lane>=16 -> {K2, pad0}
    v2f a;
    a.x = (lane < 16) ? n0 : n2;
    a.y = (lane < 16) ? n1 : 0.f;

    float s[8], w[8];
#pragma unroll
    for (int v = 0; v < 8; ++v) { s[v] = 0.f; w[v] = 0.f; }

    const float lanef = (float)(lane & 15);
    for (int c = 0; c < 625; ++c) {
        v2f b;
        b.x = B0[c * 32 + lane];
        b.y = B1[c * 32 + lane];
        v8f acc = {};
        // D = A(16x4) x B(4x16) : scores for 16 pixels x 16 bins
        acc = __builtin_amdgcn_wmma_f32_16x16x4_f32(
            /*neg_a=*/false, a, /*neg_b=*/false, b,
            /*c_mod=*/(short)0, acc, /*reuse_a=*/false, /*reuse_b=*/false);
        const float nb = (float)(c * 16) + lanef;    // bin index for this lane (same for all v)
#pragma unroll
        for (int v = 0; v < 8; ++v) {
            float sc = acc[v];
            sc = fminf(fmaxf(sc, -60.f), 60.f);      // softmax shift-invariant; clamp = overflow guard
            const float e = __expf(sc);
            s[v] += e;
            w[v] = fmaf(e, nb, w[v]);
        }
    }

    // reduce across the 16 bins held by each 16-lane half (xor stays in-group)
#pragma unroll
    for (int msk = 1; msk < 16; msk <<= 1) {
#pragma unroll
        for (int v = 0; v < 8; ++v) {
            s[v] += __shfl_xor(s[v], msk, 32);
            w[v] += __shfl_xor(w[v], msk, 32);
        }
    }

    if ((lane & 15) == 0) {
        const int off = base + ((lane >> 4) << 3);   // lane0 -> rows 0..7, lane16 -> rows 8..15
#pragma unroll
        for (int v = 0; v < 8; ++v)
            out[off + v] = w[v] / s[v];              // * (D_MAX/N_SAMP) == 1.0
    }
}

extern "C" void kernel_launch(void* const* d_in, const int* in_sizes, int n_in,
                              void* d_out, int out_size, void* d_ws, size_t ws_size,
                              hipStream_t stream) {
    (void)in_sizes; (void)n_in; (void)out_size; (void)ws_size;
    const float* gt  = (const float*)d_in[0];   // [1,128,128]
    const float* Mod = (const float*)d_in[1];   // [NS,3]
    const float* Dem = (const float*)d_in[2];   // [NS,3]
    float* ws  = (float*)d_ws;                  // needs ~320 KB
    float* out = (float*)d_out;                 // [1,128,128]

    colsum_kernel <<<3,   256, 0, stream>>>(Mod, Dem, ws);
    corr_kernel   <<<40,  256, 0, stream>>>(Mod, Dem, ws);
    stats_kernel  <<<1,   256, 0, stream>>>(ws);
    build_b_kernel<<<79,  256, 0, stream>>>(ws);
    decode_kernel <<<128, 256, 0, stream>>>(gt, ws, out);
}